// LocalAggregator_43130061586484
// MI455X (gfx1250) — compile-verified
//
#include <hip/hip_runtime.h>
#include <hip/hip_bf16.h>
#include <cstdint>

typedef __attribute__((ext_vector_type(2))) float v2f;
typedef __attribute__((ext_vector_type(8))) float v8f;

#define P_N 10000
#define G_N 6400
#define C_N 18
#define TILES (P_N / 16)     // 625 (P divisible by 16)
#define WPB 8                // waves per block (256 threads)

// ---------------------------------------------------------------------------
// Preprocess: pack per-gaussian data into 16 floats (64B) for clean float4 loads
//   q0: mu.x, mu.y, mu.z, opacity
//   q1: cxx, cyy, czz, cxy
//   q2: cyz, cxz, bitcast(mi.x), bitcast(mi.y)
//   q3: bitcast(mi.z), bitcast(radius), 0, 0
// ---------------------------------------------------------------------------
__global__ void pack_gauss_kernel(const float* __restrict__ means,
                                  const float* __restrict__ opac,
                                  const float* __restrict__ scales,
                                  const float* __restrict__ cov,
                                  float* __restrict__ pack) {
    int g = blockIdx.x * blockDim.x + threadIdx.x;
    if (g >= G_N) return;
    float mx = means[g * 3 + 0], my = means[g * 3 + 1], mz = means[g * 3 + 2];
    float sx = scales[g * 3 + 0], sy = scales[g * 3 + 1], sz = scales[g * 3 + 2];
    float smax = fmaxf(sx, fmaxf(sy, sz));
    int radius = (int)ceilf(smax * 3.0f / 0.5f);
    // voxel coords: (x - pc_min) / 0.5 truncated toward zero (in-range, non-negative)
    int mix = (int)((mx + 50.0f) * 2.0f);
    int miy = (int)((my + 50.0f) * 2.0f);
    int miz = (int)((mz + 5.0f) * 2.0f);
    const float* cg = cov + (size_t)g * 9;
    float4* dst = (float4*)(pack + (size_t)g * 16);
    float4 q0 = make_float4(mx, my, mz, opac[g]);
    float4 q1 = make_float4(cg[0], cg[4], cg[8], cg[1]);   // xx, yy, zz, xy
    float4 q2 = make_float4(cg[5], cg[2],                  // yz, xz
                            __int_as_float(mix), __int_as_float(miy));
    float4 q3 = make_float4(__int_as_float(miz), __int_as_float(radius), 0.0f, 0.0f);
    dst[0] = q0; dst[1] = q1; dst[2] = q2; dst[3] = q3;
}

// Pad semantics [G,18] -> [G,32] with zeros (second WMMA tile covers classes 16..31)
__global__ void pad_sem_kernel(const float* __restrict__ sem, float* __restrict__ semp) {
    int i = blockIdx.x * blockDim.x + threadIdx.x;
    if (i >= G_N * 32) return;
    int g = i >> 5, c = i & 31;
    semp[i] = (c < C_N) ? sem[g * C_N + c] : 0.0f;
}

// Per-pair weight: voxel-box masked exp of Mahalanobis power times opacity.
// Select, not branch: EXEC stays all-ones for WMMA.
__device__ __forceinline__ float gauss_weight(float px, float py, float pz,
                                              int pix, int piy, int piz,
                                              float4 q0, float4 q1, float4 q2, float4 q3) {
    float dx = px - q0.x, dy = py - q0.y, dz = pz - q0.z;
    float pw = -0.5f * (q1.x * dx * dx + q1.y * dy * dy + q1.z * dz * dz)
               - q1.w * dx * dy - q2.x * dy * dz - q2.y * dx * dz;
    int ax = pix - __float_as_int(q2.z); ax = ax < 0 ? -ax : ax;
    int ay = piy - __float_as_int(q2.w); ay = ay < 0 ? -ay : ay;
    int az = piz - __float_as_int(q3.x); az = az < 0 ? -az : az;
    int rad = __float_as_int(q3.y);
    bool in = (ax <= rad) && (ay <= rad) && (az <= rad);
    float w = __expf(pw) * q0.w;
    return in ? w : 0.0f;
}

// ---------------------------------------------------------------------------
// Main: one wave per 16-point tile, one G-slice per blockIdx.y.
// Per K=4 chunk: lane (hi = lane>=16) computes weights for gaussians
// gA = g0+2*hi and gA+1 -> A-matrix layout of V_WMMA_F32_16X16X4_F32.
// B vgpr r holds sem[K = g0 + r + 2*hi][class = lane%16 (+16 for acc1)].
// ---------------------------------------------------------------------------
__global__ void __launch_bounds__(256) agg_wmma_kernel(
        const float* __restrict__ pts,
        const float* __restrict__ pack,
        const float* __restrict__ semp,
        float* __restrict__ dst,
        int gPerSlice, int sliceStride) {
    int wave = threadIdx.x >> 5;
    int lane = threadIdx.x & 31;
    int tile = blockIdx.x * WPB + wave;
    if (tile >= TILES) return;                 // wave-uniform exit
    int slice = blockIdx.y;
    float* out = dst + (size_t)slice * (size_t)sliceStride;
    int gBeg = slice * gPerSlice;
    int gEnd = gBeg + gPerSlice;

    int m = lane & 15;                         // point row within tile / class col
    int hi = lane >> 4;
    int p = tile * 16 + m;
    float px = pts[p * 3 + 0], py = pts[p * 3 + 1], pz = pts[p * 3 + 2];
    int pix = (int)((px + 50.0f) * 2.0f);
    int piy = (int)((py + 50.0f) * 2.0f);
    int piz = (int)((pz + 5.0f) * 2.0f);

    v8f acc0 = {0.f, 0.f, 0.f, 0.f, 0.f, 0.f, 0.f, 0.f};  // classes 0..15
    v8f acc1 = {0.f, 0.f, 0.f, 0.f, 0.f, 0.f, 0.f, 0.f};  // classes 16..31 (16,17 real)

    for (int g0 = gBeg; g0 < gEnd; g0 += 4) {
        int gA = g0 + 2 * hi;
        const float4* qa = (const float4*)(pack + (size_t)gA * 16);
        float4 a0 = qa[0], a1 = qa[1], a2 = qa[2], a3 = qa[3];
        const float4* qb = (const float4*)(pack + (size_t)(gA + 1) * 16);
        float4 b0 = qb[0], b1 = qb[1], b2 = qb[2], b3 = qb[3];

        float wA = gauss_weight(px, py, pz, pix, piy, piz, a0, a1, a2, a3);
        float wB = gauss_weight(px, py, pz, pix, piy, piz, b0, b1, b2, b3);

        v2f A; A.x = wA; A.y = wB;
        v2f Blo, Bhi;
        const float* srowA = semp + (size_t)gA * 32;
        const float* srowB = semp + (size_t)(gA + 1) * 32;
        Blo.x = srowA[m];        Blo.y = srowB[m];
        Bhi.x = srowA[m + 16];   Bhi.y = srowB[m + 16];

        acc0 = __builtin_amdgcn_wmma_f32_16x16x4_f32(false, A, false, Blo,
                                                     (short)0, acc0, false, false);
        acc1 = __builtin_amdgcn_wmma_f32_16x16x4_f32(false, A, false, Bhi,
                                                     (short)0, acc1, false, false);
    }

    // D layout: vgpr r, lane l -> (M = r + 8*(l>=16), N = l%16)
#pragma unroll
    for (int r = 0; r < 8; ++r) {
        int prow = tile * 16 + r + 8 * hi;
        out[(size_t)prow * C_N + m] = acc0[r];
    }
    if (m < 2) {
#pragma unroll
        for (int r = 0; r < 8; ++r) {
            int prow = tile * 16 + r + 8 * hi;
            out[(size_t)prow * C_N + 16 + m] = acc1[r];
        }
    }
}

// Deterministic cross-slice reduction (no float atomics)
__global__ void reduce_kernel(const float* __restrict__ part, float* __restrict__ out, int ns) {
    int i = blockIdx.x * blockDim.x + threadIdx.x;
    if (i >= P_N * C_N) return;
    float s = 0.0f;
    for (int k = 0; k < ns; ++k) s += part[(size_t)k * (P_N * C_N) + i];
    out[i] = s;
}

// Scalar fallback (only if workspace is implausibly small)
__global__ void fallback_kernel(const float* __restrict__ pts, const float* __restrict__ means,
                                const float* __restrict__ opac, const float* __restrict__ sem,
                                const float* __restrict__ scales, const float* __restrict__ cov,
                                float* __restrict__ out) {
    int p = blockIdx.x * blockDim.x + threadIdx.x;
    if (p >= P_N) return;
    float px = pts[p * 3], py = pts[p * 3 + 1], pz = pts[p * 3 + 2];
    int pix = (int)((px + 50.f) * 2.f), piy = (int)((py + 50.f) * 2.f), piz = (int)((pz + 5.f) * 2.f);
    float acc[C_N];
    for (int c = 0; c < C_N; ++c) acc[c] = 0.0f;
    for (int g = 0; g < G_N; ++g) {
        float mx = means[g * 3], my = means[g * 3 + 1], mz = means[g * 3 + 2];
        float smax = fmaxf(scales[g * 3], fmaxf(scales[g * 3 + 1], scales[g * 3 + 2]));
        int rad = (int)ceilf(smax * 6.0f);
        int ax = pix - (int)((mx + 50.f) * 2.f); ax = ax < 0 ? -ax : ax;
        int ay = piy - (int)((my + 50.f) * 2.f); ay = ay < 0 ? -ay : ay;
        int az = piz - (int)((mz + 5.f) * 2.f);  az = az < 0 ? -az : az;
        float dx = px - mx, dy = py - my, dz = pz - mz;
        const float* cg = cov + (size_t)g * 9;
        float pw = -0.5f * (cg[0] * dx * dx + cg[4] * dy * dy + cg[8] * dz * dz)
                   - cg[1] * dx * dy - cg[5] * dy * dz - cg[2] * dx * dz;
        float w = (ax <= rad && ay <= rad && az <= rad) ? __expf(pw) * opac[g] : 0.0f;
        for (int c = 0; c < C_N; ++c) acc[c] += w * sem[(size_t)g * C_N + c];
    }
    for (int c = 0; c < C_N; ++c) out[(size_t)p * C_N + c] = acc[c];
}

static inline size_t align256(size_t x) { return (x + 255) & ~(size_t)255; }

extern "C" void kernel_launch(void* const* d_in, const int* in_sizes, int n_in,
                              void* d_out, int out_size, void* d_ws, size_t ws_size,
                              hipStream_t stream) {
    const float* pts    = (const float*)d_in[0];
    const float* means  = (const float*)d_in[1];
    const float* opac   = (const float*)d_in[2];
    const float* sem    = (const float*)d_in[3];
    const float* scales = (const float*)d_in[4];
    const float* cov    = (const float*)d_in[5];
    float* out = (float*)d_out;
    char*  ws  = (char*)d_ws;

    const size_t packBytes = (size_t)G_N * 16 * sizeof(float);   // 409,600
    const size_t sempBytes = (size_t)G_N * 32 * sizeof(float);   // 819,200
    const size_t off_pack = 0;
    const size_t off_semp = align256(off_pack + packBytes);
    const size_t off_part = align256(off_semp + sempBytes);
    const int NS_FULL = 8;                                       // 6400/8=800, %4==0
    const size_t partBytes = (size_t)NS_FULL * P_N * C_N * sizeof(float);

    if (ws_size < off_part) {  // can't even hold packed tables: scalar fallback
        fallback_kernel<<<(P_N + 255) / 256, 256, 0, stream>>>(pts, means, opac, sem,
                                                               scales, cov, out);
        return;
    }

    float* pack = (float*)(ws + off_pack);
    float* semp = (float*)(ws + off_semp);
    float* part = (float*)(ws + off_part);
    const int NS = (ws_size >= off_part + partBytes) ? NS_FULL : 1;

    pack_gauss_kernel<<<(G_N + 255) / 256, 256, 0, stream>>>(means, opac, scales, cov, pack);
    pad_sem_kernel<<<(G_N * 32 + 255) / 256, 256, 0, stream>>>(sem, semp);

    dim3 grid((TILES + WPB - 1) / WPB, NS);
    if (NS > 1) {
        agg_wmma_kernel<<<grid, 256, 0, stream>>>(pts, pack, semp, part,
                                                  G_N / NS, P_N * C_N);
        reduce_kernel<<<(P_N * C_N + 255) / 256, 256, 0, stream>>>(part, out, NS);
    } else {
        agg_wmma_kernel<<<grid, 256, 0, stream>>>(pts, pack, semp, out, G_N, 0);
    }
}